// HSEGNNFlexLayer_81844896793191
// MI455X (gfx1250) — compile-verified
//
#include <hip/hip_runtime.h>
#include <hip/hip_bf16.h>
#include <stddef.h>
#include <stdint.h>

typedef __attribute__((ext_vector_type(2))) float v2f;
typedef __attribute__((ext_vector_type(8))) float v8f;
typedef __attribute__((ext_vector_type(4))) int v4i;

#define D_ 128
#define A_ 8
#define AM_ 3
#define PSTRIDE 260  // 259 used cols + 1 pad; row stride = 1040 B (16B aligned)
#define TILEF (16 * PSTRIDE)

typedef __attribute__((address_space(1))) v4i as1_v4i;
typedef __attribute__((address_space(3))) v4i as3_v4i;

__device__ inline void lds_fence() {
  asm volatile("s_wait_dscnt 0" ::: "memory");
}
__device__ inline void async_wait() {
  asm volatile("s_wait_asynccnt 0" ::: "memory");
}

__device__ inline void atomic_add_f32_dev(float* p, float v) {
  asm volatile("global_atomic_add_f32 %0, %1, off scope:SCOPE_DEV"
               :: "v"(p), "v"(v) : "memory");
}

// 16B global -> LDS copy; async path (ASYNCcnt-tracked) if builtin available.
__device__ inline void async_copy16(const float* gp, float* lp) {
#if defined(__has_builtin)
#if __has_builtin(__builtin_amdgcn_global_load_async_to_lds_b128)
  __builtin_amdgcn_global_load_async_to_lds_b128(
      (as1_v4i*)(uintptr_t)gp,
      (as3_v4i*)(unsigned)(uintptr_t)lp, 0, 0);
  return;
#endif
#endif
  *(float4*)lp = *(const float4*)gp;
}

__device__ inline float swishf(float v) { return v / (1.0f + __expf(-v)); }

// One bilinear-TP layer as a GEMM over flattened K = 8*IN.
// A'[row, 8*i+k] = panel[row][i]*attr[row][k]; B' = W as [8*IN,128] (row 8i+k).
// Lane roles: r=lane&15, half=lane>>4, kb=2*half. WMMA f32 16x16x4 A layout:
// VGPR0 = K-slot0/slot2 (lane halves), VGPR1 = slot1/slot3.
// Flat rows per step i, parity p: 8i + 4p + kb + {0,1}.
// PACKED: Wp[i][t][lane] = float4{W[8i+kb][c], W[8i+kb+1][c], W[8i+kb+4][c], W[8i+kb+5][c]},
// c = t*16+r  -> one coalesced b128 per (i,t) instead of 4 b32 loads.
template <int MT, bool PACKED>
__device__ inline void tp_gemm(const float* __restrict__ panel, int r, int lane,
                               int kb, const float (&ea)[MT][4],
                               const float* __restrict__ W,
                               const float4* __restrict__ Wp, int IN,
                               v8f (&acc)[MT][8]) {
#pragma clang loop unroll(disable)
  for (int i = 0; i < IN; ++i) {
    float m[MT];
    v2f a0[MT], a1[MT];
#pragma unroll
    for (int mt = 0; mt < MT; ++mt) {
      m[mt] = panel[mt * TILEF + r * PSTRIDE + i];
      a0[mt][0] = m[mt] * ea[mt][0];
      a0[mt][1] = m[mt] * ea[mt][1];
      a1[mt][0] = m[mt] * ea[mt][2];
      a1[mt][1] = m[mt] * ea[mt][3];
    }
#pragma unroll
    for (int t = 0; t < 8; ++t) {
      float4 w;
      if (PACKED) {
        w = Wp[((size_t)i * 8 + t) * 32 + lane];
      } else {
        const float* wrow = W + (size_t)(8 * i + kb) * D_ + t * 16 + r;
        w = float4{wrow[0], wrow[128], wrow[512], wrow[640]};
      }
      v2f bf0; bf0[0] = w.x; bf0[1] = w.y;
      v2f bf1; bf1[0] = w.z; bf1[1] = w.w;
#pragma unroll
      for (int mt = 0; mt < MT; ++mt) {
        acc[mt][t] = __builtin_amdgcn_wmma_f32_16x16x4_f32(
            false, a0[mt], false, bf0, (short)0, acc[mt][t], false, false);
        acc[mt][t] = __builtin_amdgcn_wmma_f32_16x16x4_f32(
            false, a1[mt], false, bf1, (short)0, acc[mt][t], false, false);
      }
    }
  }
}

__global__ void __launch_bounds__(64) hsegnn_zero_kernel(float4* p, long long n4) {
  long long i = (long long)blockIdx.x * blockDim.x + threadIdx.x;
  if (i < n4) p[i] = float4{0.f, 0.f, 0.f, 0.f};
}

// Repack W (IN,8,128) into WMMA-B-fragment-major float4 tiles (see tp_gemm).
__global__ void __launch_bounds__(256) hsegnn_repack_kernel(
    const float* __restrict__ W, float4* __restrict__ Wp, int IN) {
  const int idx = blockIdx.x * blockDim.x + threadIdx.x;
  if (idx >= IN * 256) return;
  const int lane = idx & 31, t = (idx >> 5) & 7, i = idx >> 8;
  const int r = lane & 15, half = lane >> 4, kb = half * 2;
  const float* wrow = W + (size_t)(8 * i + kb) * D_ + t * 16 + r;
  Wp[idx] = float4{wrow[0], wrow[128], wrow[512], wrow[640]};
}

// Edge stage: M=32 per wave (two 16-row tiles share every B-fragment).
template <bool PACKED>
__global__ void __launch_bounds__(32) hsegnn_edge_kernel(
    const float* __restrict__ x, const int* __restrict__ eidx,
    const float* __restrict__ eattr, const float* __restrict__ amf,
    const float* __restrict__ W1, const float* __restrict__ bias1,
    const float* __restrict__ W2, const float* __restrict__ bias2,
    const float4* __restrict__ Wp1, const float4* __restrict__ Wp2,
    float* __restrict__ agg, int E, int nTiles) {
  __shared__ __align__(16) float panel[2 * TILEF];
  const int lane = threadIdx.x & 31;
  const int tile = blockIdx.x;
  if (tile >= nTiles) return;
  const int e0 = tile * 32;
  const int r = lane & 15, half = lane >> 4, kb = half * 2;

  // ---- gather x[dst] | x[src] | amf into the two LDS A-panels ----
  for (int rr = 0; rr < 32; ++rr) {
    const int e = e0 + rr;
    const int di = eidx[E + e];  // dst (i)
    const int si = eidx[e];      // src (j)
    float* prow = panel + (rr >> 4) * TILEF + (rr & 15) * PSTRIDE;
    async_copy16(x + (size_t)di * D_ + lane * 4, prow + lane * 4);
    async_copy16(x + (size_t)si * D_ + lane * 4, prow + D_ + lane * 4);
  }
  if (lane < 16) {
#pragma unroll
    for (int mt = 0; mt < 2; ++mt) {
      const int e = e0 + mt * 16 + lane;
      float* prow = panel + mt * TILEF + lane * PSTRIDE;
      prow[2 * D_ + 0] = amf[(size_t)e * AM_ + 0];
      prow[2 * D_ + 1] = amf[(size_t)e * AM_ + 1];
      prow[2 * D_ + 2] = amf[(size_t)e * AM_ + 2];
    }
  }
  async_wait();
  lds_fence();

  float ea[2][4];
#pragma unroll
  for (int mt = 0; mt < 2; ++mt) {
    const float* bap = eattr + (size_t)(e0 + mt * 16 + r) * A_ + kb;
    ea[mt][0] = bap[0]; ea[mt][1] = bap[1];
    ea[mt][2] = bap[4]; ea[mt][3] = bap[5];
  }

  const v8f zero8 = {0, 0, 0, 0, 0, 0, 0, 0};
  v8f acc[2][8];
#pragma unroll
  for (int mt = 0; mt < 2; ++mt)
#pragma unroll
    for (int t = 0; t < 8; ++t) acc[mt][t] = zero8;

  tp_gemm<2, PACKED>(panel, r, lane, kb, ea, W1, Wp1, 2 * D_ + AM_, acc);

  lds_fence();
#pragma unroll
  for (int mt = 0; mt < 2; ++mt)
#pragma unroll
    for (int t = 0; t < 8; ++t) {
      const float bi = bias1[t * 16 + r];
#pragma unroll
      for (int j = 0; j < 8; ++j)
        panel[mt * TILEF + (j + 8 * half) * PSTRIDE + t * 16 + r] =
            swishf(acc[mt][t][j] + bi);
    }
  lds_fence();

#pragma unroll
  for (int mt = 0; mt < 2; ++mt)
#pragma unroll
    for (int t = 0; t < 8; ++t) acc[mt][t] = zero8;

  tp_gemm<2, PACKED>(panel, r, lane, kb, ea, W2, Wp2, D_, acc);

  int nd[2][8];
#pragma unroll
  for (int mt = 0; mt < 2; ++mt)
#pragma unroll
    for (int j = 0; j < 8; ++j)
      nd[mt][j] = eidx[E + e0 + mt * 16 + j + 8 * half];
#pragma unroll
  for (int mt = 0; mt < 2; ++mt)
#pragma unroll
    for (int t = 0; t < 8; ++t) {
      const float bi = bias2[t * 16 + r];
#pragma unroll
      for (int j = 0; j < 8; ++j) {
        const float v = swishf(acc[mt][t][j] + bi);
        atomic_add_f32_dev(agg + (size_t)nd[mt][j] * D_ + t * 16 + r, v);
      }
    }
}

// Node stage: M=16 per wave, 2 waves/block. agg lives in `out` on entry.
template <bool PACKED>
__global__ void __launch_bounds__(64) hsegnn_node_kernel(
    const float* __restrict__ x, const float* __restrict__ nattr,
    const float* __restrict__ anf,
    const float* __restrict__ W3, const float* __restrict__ bias3,
    const float* __restrict__ W4, const float* __restrict__ bias4,
    const float4* __restrict__ Wp3, const float4* __restrict__ Wp4,
    float* __restrict__ out, int N, int nTiles) {
  __shared__ __align__(16) float panelS[2][TILEF];
  const int wave = threadIdx.x >> 5;
  const int lane = threadIdx.x & 31;
  const int tile = blockIdx.x * 2 + wave;
  if (tile >= nTiles) return;
  const int n0 = tile * 16;
  float* panel = panelS[wave];
  const int r = lane & 15, half = lane >> 4, kb = half * 2;

  for (int rr = 0; rr < 16; ++rr) {
    const int n = n0 + rr;
    float* prow = panel + rr * PSTRIDE;
    async_copy16(x + (size_t)n * D_ + lane * 4, prow + lane * 4);
    async_copy16(out + (size_t)n * D_ + lane * 4, prow + D_ + lane * 4);  // agg
  }
  if (lane < 16) {
    const int n = n0 + lane;
    float* prow = panel + lane * PSTRIDE;
    prow[2 * D_ + 0] = anf[(size_t)n * AM_ + 0];
    prow[2 * D_ + 1] = anf[(size_t)n * AM_ + 1];
    prow[2 * D_ + 2] = anf[(size_t)n * AM_ + 2];
  }
  async_wait();
  lds_fence();

  float na[1][4];
  {
    const float* bap = nattr + (size_t)(n0 + r) * A_ + kb;
    na[0][0] = bap[0]; na[0][1] = bap[1];
    na[0][2] = bap[4]; na[0][3] = bap[5];
  }

  const v8f zero8 = {0, 0, 0, 0, 0, 0, 0, 0};
  v8f acc[1][8];
#pragma unroll
  for (int t = 0; t < 8; ++t) acc[0][t] = zero8;

  tp_gemm<1, PACKED>(panel, r, lane, kb, na, W3, Wp3, 2 * D_ + AM_, acc);

  lds_fence();
#pragma unroll
  for (int t = 0; t < 8; ++t) {
    const float bi = bias3[t * 16 + r];
#pragma unroll
    for (int j = 0; j < 8; ++j)
      panel[(j + 8 * half) * PSTRIDE + t * 16 + r] = swishf(acc[0][t][j] + bi);
  }
  lds_fence();

#pragma unroll
  for (int t = 0; t < 8; ++t) acc[0][t] = zero8;

  tp_gemm<1, PACKED>(panel, r, lane, kb, na, W4, Wp4, D_, acc);

#pragma unroll
  for (int t = 0; t < 8; ++t) {
    const float bi = bias4[t * 16 + r];
#pragma unroll
    for (int j = 0; j < 8; ++j)
      out[(size_t)(n0 + j + 8 * half) * D_ + t * 16 + r] = acc[0][t][j] + bi;
  }
}

extern "C" void kernel_launch(void* const* d_in, const int* in_sizes, int n_in,
                              void* d_out, int out_size, void* d_ws, size_t ws_size,
                              hipStream_t stream) {
  (void)n_in; (void)out_size;
  const float* x     = (const float*)d_in[0];
  const int*   eidx  = (const int*)d_in[1];    // [2, E]
  const float* eattr = (const float*)d_in[2];  // [E, 8]
  const float* nattr = (const float*)d_in[3];  // [N, 8]
  /* d_in[4] = batch (unused) */
  const float* amf   = (const float*)d_in[5];  // [E, 3]
  const float* anf   = (const float*)d_in[6];  // [N, 3]
  const float* W1 = (const float*)d_in[7];
  const float* b1 = (const float*)d_in[8];
  const float* W2 = (const float*)d_in[9];
  const float* b2 = (const float*)d_in[10];
  const float* W3 = (const float*)d_in[11];
  const float* b3 = (const float*)d_in[12];
  const float* W4 = (const float*)d_in[13];
  const float* b4 = (const float*)d_in[14];
  float* out = (float*)d_out;

  const int N = in_sizes[0] / D_;  // 50000
  const int E = in_sizes[2] / A_;  // 500000
  const int IN1 = 2 * D_ + AM_;    // 259

  // Packed-weight layout in d_ws (float4 units)
  const size_t f4a = (size_t)IN1 * 8 * 32;  // W1/W3: 66304 float4
  const size_t f4b = (size_t)D_ * 8 * 32;   // W2/W4: 32768 float4
  const size_t needBytes = (2 * f4a + 2 * f4b) * sizeof(float4);  // ~3.2 MB
  const bool packed = (d_ws != nullptr) && (ws_size >= needBytes);
  float4* Wp1 = (float4*)d_ws;
  float4* Wp2 = Wp1 + f4a;
  float4* Wp3 = Wp2 + f4b;
  float4* Wp4 = Wp3 + f4a;

  // 1) zero the aggregation buffer (== d_out)
  const long long n4 = (long long)N * (D_ / 4);
  hsegnn_zero_kernel<<<(unsigned)((n4 + 63) / 64), 64, 0, stream>>>((float4*)out, n4);

  // 2) repack weights for coalesced b128 B-fragment loads
  if (packed) {
    hsegnn_repack_kernel<<<(unsigned)IN1, 256, 0, stream>>>(W1, Wp1, IN1);
    hsegnn_repack_kernel<<<(unsigned)D_, 256, 0, stream>>>(W2, Wp2, D_);
    hsegnn_repack_kernel<<<(unsigned)IN1, 256, 0, stream>>>(W3, Wp3, IN1);
    hsegnn_repack_kernel<<<(unsigned)D_, 256, 0, stream>>>(W4, Wp4, D_);
  }

  // 3) edge stage (message + scatter-add); E % 32 == 0 for given shapes
  const int eTiles = E / 32;
  if (packed)
    hsegnn_edge_kernel<true><<<(unsigned)eTiles, 32, 0, stream>>>(
        x, eidx, eattr, amf, W1, b1, W2, b2, Wp1, Wp2, out, E, eTiles);
  else
    hsegnn_edge_kernel<false><<<(unsigned)eTiles, 32, 0, stream>>>(
        x, eidx, eattr, amf, W1, b1, W2, b2, nullptr, nullptr, out, E, eTiles);

  // 4) node stage (update): reads agg from `out`, overwrites with final result
  const int nTiles = N / 16;
  if (packed)
    hsegnn_node_kernel<true><<<(unsigned)((nTiles + 1) / 2), 64, 0, stream>>>(
        x, nattr, anf, W3, b3, W4, b4, Wp3, Wp4, out, N, nTiles);
  else
    hsegnn_node_kernel<false><<<(unsigned)((nTiles + 1) / 2), 64, 0, stream>>>(
        x, nattr, anf, W3, b3, W4, b4, nullptr, nullptr, out, N, nTiles);
}